// LatentGP_59768764891677
// MI455X (gfx1250) — compile-verified
//
#include <hip/hip_runtime.h>

typedef float v2f __attribute__((ext_vector_type(2)));
typedef float v8f __attribute__((ext_vector_type(8)));

#define LDP 132   // padded LDS pitch: 528B rows -> 16B-aligned for async B128 loads

// ---------------------------------------------------------------------------
// Async global->LDS tile streamer (CDNA5 GLOBAL_LOAD_ASYNC_TO_LDS_B128).
// One wave row-streams 512B rows: 32 lanes x 16B. No VGPR staging.
// Caller must issue ASYNC_TILE_WAIT() then __syncthreads() before use.
// ---------------------------------------------------------------------------
__device__ __forceinline__ void async_tile_load(unsigned lds_base_bytes,
                                                const float* __restrict__ gbase,
                                                int tid, int nthreads) {
  int lane = tid & 31;
  int wid = tid >> 5;
  int nw = nthreads >> 5;
  for (int r = wid; r < 128; r += nw) {
    unsigned laddr = lds_base_bytes + (unsigned)(r * (LDP * 4)) + (unsigned)(lane << 4);
    const float* src = gbase + (size_t)r * 2048 + (lane << 2);
    asm volatile("global_load_async_to_lds_b128 %0, %1, off"
                 :: "v"(laddr), "v"(src)
                 : "memory");
  }
}

__device__ __forceinline__ void async_tile_load_k(unsigned lds_base_bytes,
                                                  const float* __restrict__ gbase,
                                                  int tid, int nthreads) {
  // same but for 128-float-pitch source (the K/B buffers)
  int lane = tid & 31;
  int wid = tid >> 5;
  int nw = nthreads >> 5;
  for (int r = wid; r < 128; r += nw) {
    unsigned laddr = lds_base_bytes + (unsigned)(r * (LDP * 4)) + (unsigned)(lane << 4);
    const float* src = gbase + (size_t)r * 128 + (lane << 2);
    asm volatile("global_load_async_to_lds_b128 %0, %1, off"
                 :: "v"(laddr), "v"(src)
                 : "memory");
  }
}

#define ASYNC_TILE_WAIT() asm volatile("s_wait_asynccnt 0x0" ::: "memory")

__device__ __forceinline__ unsigned lds_addr_bytes(const void* p) {
  return (unsigned)(size_t)p;   // LDS flat addr: low 32 bits = LDS byte offset
}

// ---------------------------------------------------------------------------
// K_d[a][b] = exp(-(T[a]-T[b])^2 / (2*ks[d]^2)),  d<16, a,b<128
// ---------------------------------------------------------------------------
__global__ void lgp_build_K(const float* __restrict__ T, const float* __restrict__ ks,
                            float* __restrict__ K) {
  int d = blockIdx.x >> 7;
  int a = blockIdx.x & 127;
  int b = threadIdx.x;
  float s = ks[d];
  float inv = 1.0f / (2.0f * s * s);
  float dt = T[a] - T[b];
  K[(d << 14) + (a << 7) + b] = expf(-dt * dt * inv);
}

// ---------------------------------------------------------------------------
// prep: xx = ||X||^2, M = A^T A (16x16), Y = X A (128x16, row-major [a][d])
// ---------------------------------------------------------------------------
__global__ void lgp_prep(const float* __restrict__ X, const float* __restrict__ A,
                         float* __restrict__ Y, float* __restrict__ M,
                         float* __restrict__ scal) {
  __shared__ float Ash[64 * 16];
  __shared__ float red[256];
  int tid = threadIdx.x;
  for (int i = tid; i < 1024; i += 256) Ash[i] = A[i];
  float acc = 0.0f;
  for (int i = tid; i < 8192; i += 256) { float v = X[i]; acc += v * v; }
  red[tid] = acc;
  __syncthreads();
  for (int s = 128; s > 0; s >>= 1) {
    if (tid < s) red[tid] += red[tid + s];
    __syncthreads();
  }
  if (tid == 0) scal[0] = red[0];
  {
    int d = tid >> 4, e = tid & 15;
    float m = 0.0f;
    for (int i = 0; i < 64; i++) m += Ash[i * 16 + d] * Ash[i * 16 + e];
    M[tid] = m;
  }
  for (int p = tid; p < 2048; p += 256) {
    int a = p >> 4, d = p & 15;
    const float* xr = X + a * 64;
    float y = 0.0f;
    for (int i = 0; i < 64; i++) y += xr[i] * Ash[i * 16 + d];
    Y[p] = y;
  }
}

// ---------------------------------------------------------------------------
// One-sided Jacobi per d: factor K_d = B_d B_d^T (PSD, rank-safe).
// In-LDS 128x128 (padded), round-robin pairing, 2 threads per column pair.
// Writes B_d in place of K_d.
// ---------------------------------------------------------------------------
__global__ void lgp_jacobi(float* __restrict__ KB) {
  extern __shared__ float G[];   // 128*LDP floats
  __shared__ float spp[128], sqq[128], spq[128];
  int d = blockIdx.x, tid = threadIdx.x;
  float* Kd = KB + (d << 14);
  async_tile_load_k(lds_addr_bytes(G), Kd, tid, 128);
  ASYNC_TILE_WAIT();
  __syncthreads();
  int j = tid >> 1, lane = tid & 1;
  int r0 = lane << 6;
  for (int sweep = 0; sweep < 12; sweep++) {
    for (int round = 0; round < 127; round++) {
      int p = (j == 0) ? 0 : 1 + (j - 1 + round) % 127;
      int q = 1 + (126 - j + round) % 127;
      float app = 0.0f, aqq = 0.0f, apq = 0.0f;
      for (int r = r0; r < r0 + 64; r++) {
        float gp = G[r * LDP + p], gq = G[r * LDP + q];
        app += gp * gp; aqq += gq * gq; apq += gp * gq;
      }
      spp[tid] = app; sqq[tid] = aqq; spq[tid] = apq;
      __syncthreads();
      app = spp[tid] + spp[tid ^ 1];
      aqq = sqq[tid] + sqq[tid ^ 1];
      apq = spq[tid] + spq[tid ^ 1];
      float c = 1.0f, sn = 0.0f;
      if (fabsf(apq) > 1e-12f * sqrtf(app * aqq) + 1e-30f) {
        float tau = (aqq - app) / (2.0f * apq);
        float t = ((tau >= 0.0f) ? 1.0f : -1.0f) / (fabsf(tau) + sqrtf(1.0f + tau * tau));
        c = 1.0f / sqrtf(1.0f + t * t);
        sn = t * c;
      }
      for (int r = r0; r < r0 + 64; r++) {
        float gp = G[r * LDP + p], gq = G[r * LDP + q];
        G[r * LDP + p] = c * gp - sn * gq;
        G[r * LDP + q] = sn * gp + c * gq;
      }
      __syncthreads();
    }
  }
  // column i: w_i = u_i * sigma_i  ->  b_i = w_i / sqrt(sigma_i), clamp tiny
  float nn = 0.0f;
  for (int r = 0; r < 128; r++) { float g = G[r * LDP + tid]; nn += g * g; }
  float sig = sqrtf(nn);
  float sc = (sig > 1e-18f) ? (1.0f / sqrtf(sig)) : 0.0f;
  __syncthreads();
  for (int r = 0; r < 128; r++) Kd[(r << 7) + tid] = G[r * LDP + tid] * sc;
}

// ---------------------------------------------------------------------------
// y2[d*128+a] = (B_d^T Y[:,d])[a]
// ---------------------------------------------------------------------------
__global__ void lgp_utx(const float* __restrict__ KB, const float* __restrict__ Y,
                        float* __restrict__ y2) {
  __shared__ float ys[128];
  int d = blockIdx.x, a = threadIdx.x;
  ys[a] = Y[(a << 4) + d];
  __syncthreads();
  const float* Bd = KB + (d << 14);
  float acc = 0.0f;
  for (int b = 0; b < 128; b++) acc += Bd[(b << 7) + a] * ys[b];
  y2[(d << 7) + a] = acc;
}

// ---------------------------------------------------------------------------
// C block (dd,ee), ee<=dd:  C = sigma2*delta*I + M[dd,ee] * (B_dd^T B_ee)
// 128x128 per block via V_WMMA_F32_16X16X4_F32; 8 waves x 8 tiles each.
// ---------------------------------------------------------------------------
__global__ void lgp_buildC(const float* __restrict__ KB, const float* __restrict__ M,
                           const float* __restrict__ noise, float* __restrict__ C) {
  extern __shared__ float lds[];
  float* Bd = lds;
  float* Be = lds + 128 * LDP;
  int idx = blockIdx.x, dd = 0;
  while (idx > dd) { idx -= dd + 1; dd++; }
  int ee = idx;
  int tid = threadIdx.x;
  async_tile_load_k(lds_addr_bytes(Bd), KB + (dd << 14), tid, 256);
  async_tile_load_k(lds_addr_bytes(Be), KB + (ee << 14), tid, 256);
  ASYNC_TILE_WAIT();
  __syncthreads();
  float mde = M[dd * 16 + ee];
  float s2 = noise[0];
  int wave = tid >> 5, lane = tid & 31;
  int mrow = wave * 16 + (lane & 15);
  int kbase = (lane >> 4) << 1;
  v8f acc[8];
  v8f zero = {0.f, 0.f, 0.f, 0.f, 0.f, 0.f, 0.f, 0.f};
#pragma unroll
  for (int t = 0; t < 8; t++) acc[t] = zero;
  for (int kb = 0; kb < 128; kb += 4) {
    v2f av;                                       // A[m][k] = Bd[k][m]
    av.x = Bd[(kb + kbase) * LDP + mrow];
    av.y = Bd[(kb + kbase + 1) * LDP + mrow];
#pragma unroll
    for (int tn = 0; tn < 8; tn++) {
      int ncol = tn * 16 + (lane & 15);
      v2f bv;                                     // B[k][n] = Be[k][n]
      bv.x = Be[(kb + kbase) * LDP + ncol];
      bv.y = Be[(kb + kbase + 1) * LDP + ncol];
      acc[tn] = __builtin_amdgcn_wmma_f32_16x16x4_f32(false, av, false, bv,
                                                      (short)0, acc[tn], false, false);
    }
  }
#pragma unroll
  for (int tn = 0; tn < 8; tn++) {
    int col = tn * 16 + (lane & 15);
#pragma unroll
    for (int g = 0; g < 8; g++) {
      int row = wave * 16 + g + ((lane >> 4) << 3);
      float v = mde * acc[tn][g];
      if (dd == ee && row == col) v += s2;
      C[(dd * 128 + row) * 2048 + ee * 128 + col] = v;
    }
  }
}

// ---------------------------------------------------------------------------
// Blocked Cholesky of C (2048, NB=128): diag factor
// ---------------------------------------------------------------------------
__global__ void lgp_potrf(float* __restrict__ C, int jb) {
  extern __shared__ float L[];   // 128*LDP
  int tid = threadIdx.x;
  int j0 = jb * 128;
  async_tile_load(lds_addr_bytes(L), C + (size_t)j0 * 2048 + j0, tid, 128);
  ASYNC_TILE_WAIT();
  __syncthreads();
  for (int k = 0; k < 128; k++) {
    if (tid == 0) L[k * LDP + k] = sqrtf(L[k * LDP + k]);
    __syncthreads();
    if (tid > k) L[tid * LDP + k] /= L[k * LDP + k];
    __syncthreads();
    if (tid > k) {
      float lk = L[tid * LDP + k];
      for (int c = k + 1; c <= tid; c++) L[tid * LDP + c] -= lk * L[c * LDP + k];
    }
    __syncthreads();
  }
  for (int r = 0; r < 128; r++) C[(j0 + r) * 2048 + j0 + tid] = L[r * LDP + tid];
}

// Panel TRSM: B = A * L^{-T}; row-parallel forward substitution (thread = row).
__global__ void lgp_trsm(float* __restrict__ C, int jb) {
  extern __shared__ float lds[];
  float* Lt = lds;
  float* Bt = lds + 128 * LDP;
  int tid = threadIdx.x;
  int i = jb + 1 + blockIdx.x;
  int j0 = jb * 128, i0 = i * 128;
  async_tile_load(lds_addr_bytes(Lt), C + (size_t)j0 * 2048 + j0, tid, 128);
  async_tile_load(lds_addr_bytes(Bt), C + (size_t)i0 * 2048 + j0, tid, 128);
  ASYNC_TILE_WAIT();
  __syncthreads();
  float* br = Bt + tid * LDP;
  for (int k = 0; k < 128; k++) {
    const float* Lk = Lt + k * LDP;
    float s = br[k];
    for (int c = 0; c < k; c++) s -= Lk[c] * br[c];
    br[k] = s / Lk[k];
  }
  __syncthreads();
  for (int r = 0; r < 128; r++) C[(i0 + r) * 2048 + j0 + tid] = Bt[r * LDP + tid];
}

// SYRK/GEMM trailing update: C[i,l] -= P_i * P_l^T via WMMA f32 16x16x4.
__global__ void lgp_syrk(float* __restrict__ C, int jb) {
  extern __shared__ float lds[];
  float* Pi = lds;
  float* Pl = lds + 128 * LDP;
  int idx = blockIdx.x, ri = 0;
  while (idx > ri) { idx -= ri + 1; ri++; }
  int rl = idx;
  int i = jb + 1 + ri, l = jb + 1 + rl;
  int tid = threadIdx.x;
  int i0 = i * 128, l0 = l * 128, j0 = jb * 128;
  async_tile_load(lds_addr_bytes(Pi), C + (size_t)i0 * 2048 + j0, tid, 256);
  async_tile_load(lds_addr_bytes(Pl), C + (size_t)l0 * 2048 + j0, tid, 256);
  ASYNC_TILE_WAIT();
  __syncthreads();
  int wave = tid >> 5, lane = tid & 31;
  int mrow = wave * 16 + (lane & 15);
  int kbase = (lane >> 4) << 1;
  v8f acc[8];
  v8f zero = {0.f, 0.f, 0.f, 0.f, 0.f, 0.f, 0.f, 0.f};
#pragma unroll
  for (int t = 0; t < 8; t++) acc[t] = zero;
  for (int kb = 0; kb < 128; kb += 4) {
    v2f av;                                       // A[m][k] = Pi[m][k]
    av.x = Pi[mrow * LDP + kb + kbase];
    av.y = Pi[mrow * LDP + kb + kbase + 1];
#pragma unroll
    for (int tn = 0; tn < 8; tn++) {
      int ncol = tn * 16 + (lane & 15);
      v2f bv;                                     // B[k][n] = Pl[n][k]
      bv.x = Pl[ncol * LDP + kb + kbase];
      bv.y = Pl[ncol * LDP + kb + kbase + 1];
      acc[tn] = __builtin_amdgcn_wmma_f32_16x16x4_f32(false, av, false, bv,
                                                      (short)0, acc[tn], false, false);
    }
  }
#pragma unroll
  for (int tn = 0; tn < 8; tn++) {
    int col = l0 + tn * 16 + (lane & 15);
#pragma unroll
    for (int g = 0; g < 8; g++) {
      int row = i0 + wave * 16 + g + ((lane >> 4) << 3);
      C[row * 2048 + col] -= acc[tn][g];
    }
  }
}

// ---------------------------------------------------------------------------
// Forward solve L q = y2; qq = ||q||^2; logdet = sum log diag(L)
// ---------------------------------------------------------------------------
__global__ void lgp_solve(const float* __restrict__ C, const float* __restrict__ y2,
                          float* __restrict__ scal) {
  __shared__ float z[2048];
  __shared__ float red[256];
  int tid = threadIdx.x;
  for (int p = tid; p < 2048; p += 256) z[p] = y2[p];
  __syncthreads();
  for (int jb = 0; jb < 16; jb++) {
    int b0 = jb * 128;
    for (int k = 0; k < 128; k++) {
      int kg = b0 + k;
      if (tid == 0) z[kg] /= C[kg * 2048 + kg];
      __syncthreads();
      float zk = z[kg];
      for (int r = kg + 1 + tid; r < b0 + 128; r += 256) z[r] -= C[r * 2048 + kg] * zk;
      __syncthreads();
    }
    for (int r = b0 + 128 + tid; r < 2048; r += 256) {
      const float* Lr = C + r * 2048 + b0;
      float acc = 0.0f;
      for (int k = 0; k < 128; k++) acc += Lr[k] * z[b0 + k];
      z[r] -= acc;
    }
    __syncthreads();
  }
  float qq = 0.0f;
  for (int p = tid; p < 2048; p += 256) qq += z[p] * z[p];
  red[tid] = qq;
  __syncthreads();
  for (int s = 128; s > 0; s >>= 1) { if (tid < s) red[tid] += red[tid + s]; __syncthreads(); }
  if (tid == 0) scal[1] = red[0];
  __syncthreads();
  float ld = 0.0f;
  for (int p = tid; p < 2048; p += 256) ld += logf(C[p * 2048 + p]);
  red[tid] = ld;
  __syncthreads();
  for (int s = 128; s > 0; s >>= 1) { if (tid < s) red[tid] += red[tid + s]; __syncthreads(); }
  if (tid == 0) scal[2] = red[0];
}

__global__ void lgp_final(const float* __restrict__ scal, const float* __restrict__ noise,
                          float* __restrict__ out) {
  float s2 = noise[0];
  float xx = scal[0], qq = scal[1], ld = scal[2];
  float quad = (xx - qq) / s2;                       // x^T Sigma^-1 x
  const float n = 8192.0f, m = 2048.0f;
  float logdet = (n - m) * logf(s2) + 2.0f * ld;     // log det Sigma
  out[0] = -0.5f * (quad + logdet + n * 1.8378770664093453f); // log(2*pi)=1.837877...
}

extern "C" void kernel_launch(void* const* d_in, const int* in_sizes, int n_in,
                              void* d_out, int out_size, void* d_ws, size_t ws_size,
                              hipStream_t stream) {
  const float* T     = (const float*)d_in[0];
  const float* X     = (const float*)d_in[1];
  const float* A     = (const float*)d_in[2];
  const float* ks    = (const float*)d_in[3];
  const float* noise = (const float*)d_in[4];
  (void)in_sizes; (void)n_in; (void)out_size; (void)ws_size;

  float* ws = (float*)d_ws;
  float* KB   = ws;                 // 16*128*128 floats: K_d, then B_d in place (1 MB)
  float* C    = KB + 262144;        // 2048*2048 floats (16 MB)
  float* Y    = C + 4194304;        // 128*16
  float* y2   = Y + 2048;           // 2048
  float* M    = y2 + 2048;          // 16*16
  float* scal = M + 256;            // xx, qq, logdet

  const size_t lds1 = (size_t)128 * LDP * sizeof(float);   //  67.6 KB
  const size_t lds2 = 2 * lds1;                            // 135 KB (within 320 KB WGP LDS)

  lgp_build_K<<<2048, 128, 0, stream>>>(T, ks, KB);
  lgp_prep<<<1, 256, 0, stream>>>(X, A, Y, M, scal);
  lgp_jacobi<<<16, 128, lds1, stream>>>(KB);
  lgp_utx<<<16, 128, 0, stream>>>(KB, Y, y2);
  lgp_buildC<<<136, 256, lds2, stream>>>(KB, M, noise, C);
  for (int j = 0; j < 16; j++) {
    lgp_potrf<<<1, 128, lds1, stream>>>(C, j);
    int r = 15 - j;
    if (r > 0) {
      lgp_trsm<<<r, 128, lds2, stream>>>(C, j);
      lgp_syrk<<<r * (r + 1) / 2, 256, lds2, stream>>>(C, j);
    }
  }
  lgp_solve<<<1, 256, 0, stream>>>(C, y2, scal);
  lgp_final<<<1, 1, 0, stream>>>(scal, noise, (float*)d_out);
}